// GNN_conv_87007447482650
// MI455X (gfx1250) — compile-verified
//
#include <hip/hip_runtime.h>
#include <math.h>

#define B_SZ    512
#define N_NODES 2000
#define N_EDGES 20000
#define F_IN    16
#define H       8
#define NUM_GEN 10
#define BN_EPS  1e-3f
#define THREADS 256
#define CHUNK   128          // 8 waves * 16 rows
#define NCHUNKS 16           // ceil(2000/128)
#define ECHUNKS 157          // ceil(20000/128)
#define NLAYERS 12
#define LSTRIDE 304          // scale16 + shift16 + W 16x16 + bias16

typedef __attribute__((ext_vector_type(2))) float v2f;
typedef __attribute__((ext_vector_type(8))) float v8f;

// LDS float offsets
#define OFF_X      0         // 16000  (node features, 2000 x 8)
#define OFF_AGG    16000     // 16000  (segment-sum accum; later reused for logits)
#define OFF_STAGE  32000     // 4096   (per-wave 2 x 16x16 staging tiles)
#define OFF_LAYERS 36096     // 3648   (12 preprocessed layers)
#define OFF_MISC   39744     // 32
#define SMEM_FLOATS 39776

struct GnnParams {
  const float* node_info;
  const float* branch_w;
  const int*   branches;
  const float* actor_b;
  const float* actor_w;
  const float* logits_b;
  const float* logits_w;
  const float* Lbeta [NLAYERS];
  const float* Lgamma[NLAYERS];
  const float* Lmean [NLAYERS];
  const float* Lvar  [NLAYERS];
  const float* Lb    [NLAYERS];
  const float* Lw    [NLAYERS];
  int          Ldin  [NLAYERS];
  float*       out;
};

__device__ __forceinline__ void wave_fence() {
  // LDS pipeline is in-order per wave; just stop the compiler/scheduler from
  // moving DS ops across this point.
  asm volatile("" ::: "memory");
  __builtin_amdgcn_wave_barrier();
  asm volatile("" ::: "memory");
}

__device__ __forceinline__ float gelu_exact(float v) {
  return 0.5f * v * (1.0f + erff(v * 0.70710678118654752440f));
}

__device__ __forceinline__ v8f wmma4(v2f a, v2f b, v8f c) {
  // V_WMMA_F32_16X16X4_F32: D(16x16,f32) = A(16x4) * B(4x16) + C
  return __builtin_amdgcn_wmma_f32_16x16x4_f32(
      false, a, false, b, (short)0, c, false, false);
}

// One BN+dense layer: reads raw staged tile sin[16][16] (row-major, stride 16),
// applies y[m][n] = sum_k (sin[m][k]*scale[k]+shift[k]) * W[k][n].
// Padded region (k>=din, n>=8) of scale/shift/W is zero, so padded outputs are 0.
__device__ __forceinline__ v8f wave_bn_gemm(const float* sin, const float* L, int kdim) {
  const float* scale = L;          // [16]
  const float* shift = L + 16;     // [16]
  const float* W     = L + 32;     // [16][16]
  const int lane  = threadIdx.x & 31;
  const int half2 = (lane >> 4) << 1;   // 0 (lanes 0-15) or 2 (lanes 16-31)
  const int idx   = lane & 15;
  v8f c = {};
  for (int k0 = 0; k0 < kdim; k0 += 4) {
    const int ka = k0 + half2;
    v2f a, bb;
    // A fragment: lane (idx, half) holds A[idx][ka], A[idx][ka+1]
    a.x = sin[idx * 16 + ka]     * scale[ka]     + shift[ka];
    a.y = sin[idx * 16 + ka + 1] * scale[ka + 1] + shift[ka + 1];
    // B fragment: column idx, rows ka / ka+1
    bb.x = W[ka * 16 + idx];
    bb.y = W[(ka + 1) * 16 + idx];
    c = wmma4(a, bb, c);
  }
  return c;
}

// D layout: VGPR r, lane l -> row r + (l>=16 ? 8 : 0), col l&15.
// Only columns n<8 are useful (cols 8..15 are zero padding never read
// downstream: the second GEMM layer reads only k<8 and all tile consumers
// read only f<8). To keep every lane busy, the 16x8 = 128 GELU evaluations
// are redistributed over all 32 lanes (4 each): lanes with bit3 set take the
// upper 4 rows of column (n-8), fetched from lane (l & ~8) via shuffle.
// This halves the serial transcendental chains per tile vs. exec-masking.
__device__ __forceinline__ void wave_store_gelu(float* sout, v8f c, const float* bias) {
  const int lane  = threadIdx.x & 31;
  const int n8    = lane & 7;                         // column 0..7
  const int rbase = ((lane & 16) >> 1) | ((lane & 8) >> 1);  // 0,4,8,12
  const int src   = lane & ~8;                        // lane owning column n8
  const float bn  = bias[n8];
#pragma unroll
  for (int j = 0; j < 4; ++j) {
    const float hi = __shfl(c[4 + j], src, 32);       // rows mh+4..mh+7 of col n8
    const float v  = (lane & 8) ? hi : c[j];
    sout[(rbase + j) * 16 + n8] = gelu_exact(v + bn);
  }
}

// Two-layer FFN (BN -> dense -> gelu) x2 on a staged 16-row tile.
// Input tile (raw, pre-BN) in s_in; result (16 x 8) left in s_in (cols 8..15
// of the result are stale garbage; no consumer reads them).
__device__ __forceinline__ void wave_ffn2(float* s_in, float* s_tmp,
                                          const float* L0, const float* L1, int din0) {
  v8f c0 = wave_bn_gemm(s_in, L0, din0);
  wave_fence();
  wave_store_gelu(s_tmp, c0, L0 + 288);
  wave_fence();
  v8f c1 = wave_bn_gemm(s_tmp, L1, 8);
  wave_fence();
  wave_store_gelu(s_in, c1, L1 + 288);
  wave_fence();
}

__device__ __forceinline__ void graph_conv(
    float* xf, float* agg, float* s_in, float* s_tmp,
    const float* Lmsg0, const float* Lmsg1,
    const float* Lemb0, const float* Lemb1,
    const int* node_idx, const int* nbr_idx, const float* bw,
    int wave, int lane, int tid) {
  // zero segment-sum accumulator
  for (int i = tid; i < N_NODES * H; i += THREADS) agg[i] = 0.f;
  __syncthreads();

  // ---- edge phase: msgs = msgFFN(x[nbr]) * w_e ; agg[node] += msgs ----
  for (int ch = 0; ch < ECHUNKS; ++ch) {
    const int ebase = ch * CHUNK + wave * 16;
    // prefetch next chunk's sequential index/weight streams into cache
    {
      const int pe = ebase + CHUNK + lane;
      if (pe < N_EDGES) {
        __builtin_prefetch(&nbr_idx[pe], 0, 3);
        __builtin_prefetch(&node_idx[pe], 0, 3);
        __builtin_prefetch(&bw[pe], 0, 3);
      }
    }
    for (int t = lane; t < 256; t += 32) {
      const int m = t >> 4, k = t & 15;
      const int e = ebase + m;
      float v = 0.f;
      if (e < N_EDGES && k < H) v = xf[nbr_idx[e] * H + k];
      s_in[m * 16 + k] = v;
    }
    wave_fence();
    wave_ffn2(s_in, s_tmp, Lmsg0, Lmsg1, 8);
    for (int t = lane; t < 128; t += 32) {
      const int m = t >> 3, f = t & 7;
      const int e = ebase + m;
      if (e < N_EDGES) {
        atomicAdd(&agg[node_idx[e] * H + f], s_in[m * 16 + f] * bw[e]);
      }
    }
  }
  __syncthreads();

  // ---- node phase: emb = embFFN(concat(x, agg)); x = l2norm(emb) + x ----
  for (int ch = 0; ch < NCHUNKS; ++ch) {
    const int base = ch * CHUNK + wave * 16;
    for (int t = lane; t < 256; t += 32) {
      const int m = t >> 4, k = t & 15;
      const int node = base + m;
      float v = 0.f;
      if (node < N_NODES) v = (k < H) ? xf[node * H + k] : agg[node * H + (k - H)];
      s_in[m * 16 + k] = v;
    }
    wave_fence();
    wave_ffn2(s_in, s_tmp, Lemb0, Lemb1, 16);
    if (lane < 16) {
      const int node = base + lane;
      if (node < N_NODES) {
        float sq = 0.f;
#pragma unroll
        for (int f = 0; f < H; ++f) { float e = s_in[lane * 16 + f]; sq += e * e; }
        const float r = rsqrtf(fmaxf(sq, 1e-12f));
#pragma unroll
        for (int f = 0; f < H; ++f) {
          xf[node * H + f] += s_in[lane * 16 + f] * r;
        }
      }
    }
    wave_fence();
  }
  __syncthreads();
}

__global__ __launch_bounds__(THREADS, 1)
void gnn_fused_kernel(GnnParams P) {
  extern __shared__ float smem[];
  float* xf    = smem + OFF_X;
  float* agg   = smem + OFF_AGG;
  float* stage = smem + OFF_STAGE;
  float* lw    = smem + OFF_LAYERS;
  float* misc  = smem + OFF_MISC;

  const int b    = blockIdx.x;
  const int tid  = threadIdx.x;
  const int wave = tid >> 5;
  const int lane = tid & 31;
  float* s_in  = stage + wave * 512;
  float* s_tmp = s_in + 256;

  // ---- load weights into LDS, folding BN into scale/shift, zero-padded ----
  for (int l = 0; l < NLAYERS; ++l) {
    float* L = lw + l * LSTRIDE;
    const int din = P.Ldin[l];
    if (tid < 16) {
      const int k = tid;
      float sc = 0.f, sh = 0.f;
      if (k < din) {
        sc = P.Lgamma[l][k] * rsqrtf(P.Lvar[l][k] + BN_EPS);
        sh = P.Lbeta[l][k] - P.Lmean[l][k] * sc;
      }
      L[k] = sc;
      L[16 + k] = sh;
      L[288 + k] = (k < H) ? P.Lb[l][k] : 0.f;
    }
    {
      const int kk = tid >> 4, nn = tid & 15;
      float w = 0.f;
      if (kk < din && nn < H) w = P.Lw[l][kk * H + nn];
      L[32 + kk * 16 + nn] = w;
    }
  }
  if (tid == 0) misc[0] = P.logits_b[0];
  if (tid < H) misc[1 + tid] = P.logits_w[tid];
  __syncthreads();

  const float* ninfo    = P.node_info + (size_t)b * N_NODES * F_IN;
  const float* bw       = P.branch_w + (size_t)b * N_EDGES;
  const int*   node_idx = P.branches;
  const int*   nbr_idx  = P.branches + N_EDGES;

  // ---- pre FFN: node_info (16) -> 8 -> 8, kept resident in LDS ----
  for (int ch = 0; ch < NCHUNKS; ++ch) {
    const int base = ch * CHUNK + wave * 16;
    for (int t = lane; t < 256; t += 32) {
      const int m = t >> 4, k = t & 15;
      const int node = base + m;
      s_in[m * 16 + k] = (node < N_NODES) ? ninfo[node * F_IN + k] : 0.f;
    }
    wave_fence();
    wave_ffn2(s_in, s_tmp, lw + 0 * LSTRIDE, lw + 1 * LSTRIDE, 16);
    for (int t = lane; t < 128; t += 32) {
      const int m = t >> 3, f = t & 7;
      const int node = base + m;
      if (node < N_NODES) xf[node * H + f] = s_in[m * 16 + f];
    }
  }
  __syncthreads();

  // ---- two graph convs ----
  graph_conv(xf, agg, s_in, s_tmp,
             lw + 2 * LSTRIDE, lw + 3 * LSTRIDE,
             lw + 4 * LSTRIDE, lw + 5 * LSTRIDE,
             node_idx, nbr_idx, bw, wave, lane, tid);
  graph_conv(xf, agg, s_in, s_tmp,
             lw + 6 * LSTRIDE, lw + 7 * LSTRIDE,
             lw + 8 * LSTRIDE, lw + 9 * LSTRIDE,
             node_idx, nbr_idx, bw, wave, lane, tid);

  // ---- post FFN + logits (8 -> 1); logits stored in agg[0..N) ----
  for (int ch = 0; ch < NCHUNKS; ++ch) {
    const int base = ch * CHUNK + wave * 16;
    for (int t = lane; t < 256; t += 32) {
      const int m = t >> 4, k = t & 15;
      const int node = base + m;
      s_in[m * 16 + k] = (node < N_NODES && k < H) ? xf[node * H + k] : 0.f;
    }
    wave_fence();
    wave_ffn2(s_in, s_tmp, lw + 10 * LSTRIDE, lw + 11 * LSTRIDE, 8);
    if (lane < 16) {
      const int node = base + lane;
      if (node < N_NODES) {
        float acc = misc[0];
#pragma unroll
        for (int f = 0; f < H; ++f) acc += s_in[lane * 16 + f] * misc[1 + f];
        agg[node] = acc;
      }
    }
    wave_fence();
  }
  __syncthreads();

  // ---- actor: (2000,) @ (2000,10), deterministic two-level reduction ----
  {
    float acc[NUM_GEN];
#pragma unroll
    for (int j = 0; j < NUM_GEN; ++j) acc[j] = 0.f;
    for (int n = tid; n < N_NODES; n += THREADS) {
      const float lg = agg[n];
      const float* wrow = P.actor_w + n * NUM_GEN;
#pragma unroll
      for (int j = 0; j < NUM_GEN; ++j) acc[j] += lg * wrow[j];
    }
#pragma unroll
    for (int j = 0; j < NUM_GEN; ++j) xf[tid * NUM_GEN + j] = acc[j];
  }
  __syncthreads();
  if (tid < NUM_GEN) {
    float acc = P.actor_b[tid];
    for (int t = 0; t < THREADS; ++t) acc += xf[t * NUM_GEN + tid];
    misc[16 + tid] = acc;
  }
  __syncthreads();

  // ---- softmax + PAD_IDX gather ----
  if (tid == 0) {
    float y[NUM_GEN + 1];
    float mx = -1e30f;
    for (int j = 0; j < NUM_GEN; ++j) { y[j] = misc[16 + j]; mx = fmaxf(mx, y[j]); }
    float s = 0.f;
    for (int j = 0; j < NUM_GEN; ++j) { y[j] = expf(y[j] - mx); s += y[j]; }
    const float inv = 1.f / s;
    for (int j = 0; j < NUM_GEN; ++j) y[j] *= inv;
    y[NUM_GEN] = 0.f;
    const int pad[24] = {0,1,10,10,10,10,2,10,10,10,10,10,3,10,4,5,10,6,10,10,7,8,9,10};
    for (int i = 0; i < 24; ++i) P.out[b * 24 + i] = y[pad[i]];
  }
}

extern "C" void kernel_launch(void* const* d_in, const int* in_sizes, int n_in,
                              void* d_out, int out_size, void* d_ws, size_t ws_size,
                              hipStream_t stream) {
  (void)in_sizes; (void)n_in; (void)d_ws; (void)ws_size; (void)out_size;
  GnnParams P;
  int i = 0;
  P.node_info = (const float*)d_in[i++];
  P.branch_w  = (const float*)d_in[i++];
  P.branches  = (const int*)d_in[i++];
  // params pytree, alphabetical: actor{b,w}, conv1{emb,msg}, conv2{emb,msg},
  // logits{b,w}, post, pre. Each FFN block: beta,gamma,mean,var,dense_b,dense_w.
  P.actor_b = (const float*)d_in[i++];
  P.actor_w = (const float*)d_in[i++];
  auto setBlock = [&](int l, int din) {
    P.Lbeta[l]  = (const float*)d_in[i++];
    P.Lgamma[l] = (const float*)d_in[i++];
    P.Lmean[l]  = (const float*)d_in[i++];
    P.Lvar[l]   = (const float*)d_in[i++];
    P.Lb[l]     = (const float*)d_in[i++];
    P.Lw[l]     = (const float*)d_in[i++];
    P.Ldin[l]   = din;
  };
  // device layer slots: 0,1=pre  2,3=conv1.msg  4,5=conv1.emb
  //                     6,7=conv2.msg  8,9=conv2.emb  10,11=post
  setBlock(4, 16); setBlock(5, 8);   // conv1.emb
  setBlock(2, 8);  setBlock(3, 8);   // conv1.msg
  setBlock(8, 16); setBlock(9, 8);   // conv2.emb
  setBlock(6, 8);  setBlock(7, 8);   // conv2.msg
  P.logits_b = (const float*)d_in[i++];
  P.logits_w = (const float*)d_in[i++];
  setBlock(10, 8); setBlock(11, 8);  // post
  setBlock(0, 16); setBlock(1, 8);   // pre
  P.out = (float*)d_out;

  const size_t smem_bytes = (size_t)SMEM_FLOATS * sizeof(float);
  hipFuncSetAttribute((const void*)gnn_fused_kernel,
                      hipFuncAttributeMaxDynamicSharedMemorySize, (int)smem_bytes);
  gnn_fused_kernel<<<B_SZ, THREADS, smem_bytes, stream>>>(P);
}